// LearnableMetricGNN_52939766890970
// MI455X (gfx1250) — compile-verified
//
#include <hip/hip_runtime.h>
#include <hip/hip_bf16.h>

typedef __attribute__((ext_vector_type(2))) float v2f;
typedef __attribute__((ext_vector_type(8))) float v8f;

#define GNN_N 50000
#define GNN_E 600000
#define GNN_DIN 128
#define GNN_DHID 256
#define GNN_DOUT 128

__device__ __forceinline__ float softplus_f(float x) {
    // numerically stable: max(x,0) + log1p(exp(-|x|))
    return fmaxf(x, 0.f) + log1pf(__expf(-fabsf(x)));
}

// deg[n] = 1.0 (self-loop weight)
__global__ void gnn_init_deg(float* __restrict__ deg, int n) {
    int i = blockIdx.x * blockDim.x + threadIdx.x;
    if (i < n) deg[i] = 1.0f;
}

// sp[e] = softplus(w[e]); deg[dst[e]] += sp[e]
__global__ void gnn_edge_sp_deg(const int* __restrict__ dst, const float* __restrict__ w,
                                float* __restrict__ sp, float* __restrict__ deg,
                                int e, int n) {
    int i = blockIdx.x * blockDim.x + threadIdx.x;
    if (i >= e) return;
    float s = softplus_f(w[i]);
    sp[i] = s;
    int d = min(max(dst[i], 0), n - 1);
    atomicAdd(&deg[d], s);
}

// in-place: deg -> rsqrt(deg) (0 if deg<=0, mirrors reference)
__global__ void gnn_dinv(float* __restrict__ deg, int n) {
    int i = blockIdx.x * blockDim.x + threadIdx.x;
    if (i >= n) return;
    float d = deg[i];
    deg[i] = (d > 0.f) ? __frsqrt_rn(d) : 0.f;
}

// in-place: norm[e] = sp[e] * dinv[src] * dinv[dst]
__global__ void gnn_norm(float* __restrict__ norm, const int* __restrict__ src,
                         const int* __restrict__ dst, const float* __restrict__ dinv,
                         int e, int n) {
    int i = blockIdx.x * blockDim.x + threadIdx.x;
    if (i >= e) return;
    int s = min(max(src[i], 0), n - 1);
    int d = min(max(dst[i], 0), n - 1);
    norm[i] = norm[i] * dinv[s] * dinv[d];
}

// T = (RELU ? relu(X) : X) @ W   -- exact f32 via V_WMMA_F32_16X16X4_F32.
// One 16x16 output tile per wave, 8 waves (256 threads) per block.
// K, NOUT compile-time: 32-bit address math, unrolled K loop.
template <int K, int NOUT, bool RELU>
__global__ __launch_bounds__(256) void gnn_gemm_wmma(
    const float* __restrict__ X, const float* __restrict__ W,
    float* __restrict__ T, int M) {
    const int lane = threadIdx.x & 31;
    const int wave = threadIdx.x >> 5;
    constexpr int colTiles = NOUT >> 4;
    const int tiles = (M >> 4) * colTiles;
    const int tile = blockIdx.x * 8 + wave;
    if (tile >= tiles) return;          // wave-uniform: EXEC stays all-ones

    const int rowTile = tile / colTiles;
    const int colTile = tile - rowTile * colTiles;
    const int row0 = rowTile << 4;
    const int col0 = colTile << 4;

    const int half = lane >> 4;         // 0: K=0,1  1: K=2,3 (ISA 16x4 f32 A layout)
    const int m    = lane & 15;
    const int kb   = half << 1;

    const float* __restrict__ arow = X + (row0 + m) * K + kb;   // A row m of tile
    const float* __restrict__ bcol = W + kb * NOUT + col0 + m;  // B column m of tile

    v8f acc = {0.f, 0.f, 0.f, 0.f, 0.f, 0.f, 0.f, 0.f};

#pragma unroll 8
    for (int k = 0; k < K; k += 4) {
        v2f a = *(const v2f*)(arow + k);            // global_load_b64
        if (RELU) { a.x = fmaxf(a.x, 0.f); a.y = fmaxf(a.y, 0.f); }
        v2f b;
        b.x = bcol[k * NOUT];
        b.y = bcol[k * NOUT + NOUT];
        // 8-arg form: (neg_a, A, neg_b, B, c_mod, C, reuse_a, reuse_b)
        acc = __builtin_amdgcn_wmma_f32_16x16x4_f32(
            false, a, false, b, (short)0, acc, false, false);
    }

    // D 16x16 f32 layout: VGPR r -> row (r + 8*half), col m
    float* __restrict__ dptr = T + (row0 + 8 * half) * NOUT + col0 + m;
#pragma unroll
    for (int r = 0; r < 8; ++r)
        dptr[r * NOUT] = acc[r];
}

// out[n][c] = b[c] + T[n][c] * dinv[n]^2   (bias + self-loop term), float4 wide
template <int DOUT>
__global__ void gnn_agg_init(float* __restrict__ out, const float* __restrict__ bias,
                             const float* __restrict__ T, const float* __restrict__ dinv,
                             int n) {
    constexpr int DOUT4 = DOUT >> 2;          // power of two -> shifts
    constexpr int SH = (DOUT4 == 64) ? 6 : 5;
    int i = blockIdx.x * blockDim.x + threadIdx.x;
    if (i >= n * DOUT4) return;
    int node = i >> SH;
    int c4 = i & (DOUT4 - 1);
    float di = dinv[node];
    float sn = di * di;
    float4 t = ((const float4*)(T + node * DOUT))[c4];
    float4 b = ((const float4*)bias)[c4];
    float4 o;
    o.x = b.x + t.x * sn; o.y = b.y + t.y * sn;
    o.z = b.z + t.z * sn; o.w = b.w + t.w * sn;
    ((float4*)(out + node * DOUT))[c4] = o;
}

// out[dst[e]] += norm[e] * T[src[e]]   (scatter-add, 4 channels / thread).
// Each edge owns DOUT/4 (= 64 or 32) consecutive threads, so a 32-lane wave
// never straddles edges: edge id is wave-uniform -> scalarize the per-edge
// loads (src/dst/norm) via readfirstlane so they lower to s_load.
template <int DOUT>
__global__ void gnn_agg_edges(const int* __restrict__ src, const int* __restrict__ dst,
                              const float* __restrict__ norm, const float* __restrict__ T,
                              float* __restrict__ out, int e, int n) {
    constexpr int DOUT4 = DOUT >> 2;
    constexpr int SH = (DOUT4 == 64) ? 6 : 5;
    int i = blockIdx.x * blockDim.x + threadIdx.x;
    if (i >= e * DOUT4) return;
    int edge = __builtin_amdgcn_readfirstlane(i >> SH);   // wave-uniform
    int c4 = i & (DOUT4 - 1);
    int s = min(max(src[edge], 0), n - 1);
    int d = min(max(dst[edge], 0), n - 1);
    float nv = norm[edge];
    float4 msg = ((const float4*)(T + s * DOUT))[c4];
    float* o = out + d * DOUT + (c4 << 2);
    atomicAdd(o + 0, nv * msg.x);
    atomicAdd(o + 1, nv * msg.y);
    atomicAdd(o + 2, nv * msg.z);
    atomicAdd(o + 3, nv * msg.w);
}

extern "C" void kernel_launch(void* const* d_in, const int* in_sizes, int n_in,
                              void* d_out, int out_size, void* d_ws, size_t ws_size,
                              hipStream_t stream) {
    const float* x  = (const float*)d_in[0];
    const int*   ei = (const int*)d_in[1];
    const float* ew = (const float*)d_in[2];
    const float* W1 = (const float*)d_in[3];
    const float* b1 = (const float*)d_in[4];
    const float* W2 = (const float*)d_in[5];
    const float* b2 = (const float*)d_in[6];
    const float* W3 = (const float*)d_in[7];
    const float* b3 = (const float*)d_in[8];
    float* out = (float*)d_out;

    const int* src = ei;            // edge_index[0]
    const int* dst = ei + GNN_E;    // edge_index[1]

    // Workspace carve-up (256B aligned): dinv[N], norm[E], T[N*256], A[N*256]
    char* ws = (char*)d_ws;
    auto align256 = [](size_t v) { return (v + 255) & ~(size_t)255; };
    float* dinv = (float*)ws;  ws += align256((size_t)GNN_N * 4);
    float* norm = (float*)ws;  ws += align256((size_t)GNN_E * 4);
    float* T    = (float*)ws;  ws += align256((size_t)GNN_N * GNN_DHID * 4);
    float* A    = (float*)ws;  ws += align256((size_t)GNN_N * GNN_DHID * 4);

    const int TB = 256;
    auto cdiv = [](int a, int b) { return (a + b - 1) / b; };

    // --- edge normalization preamble ---
    gnn_init_deg<<<cdiv(GNN_N, TB), TB, 0, stream>>>(dinv, GNN_N);
    gnn_edge_sp_deg<<<cdiv(GNN_E, TB), TB, 0, stream>>>(dst, ew, norm, dinv, GNN_E, GNN_N);
    gnn_dinv<<<cdiv(GNN_N, TB), TB, 0, stream>>>(dinv, GNN_N);
    gnn_norm<<<cdiv(GNN_E, TB), TB, 0, stream>>>(norm, src, dst, dinv, GNN_E, GNN_N);

    // --- layer 1: T = x @ W1 ; A = agg(T) + b1  (relu deferred into next GEMM) ---
    {
        int tiles = (GNN_N / 16) * (GNN_DHID / 16);
        gnn_gemm_wmma<GNN_DIN, GNN_DHID, false><<<cdiv(tiles, 8), 256, 0, stream>>>(x, W1, T, GNN_N);
        gnn_agg_init<GNN_DHID><<<cdiv(GNN_N * (GNN_DHID / 4), TB), TB, 0, stream>>>(A, b1, T, dinv, GNN_N);
        gnn_agg_edges<GNN_DHID><<<cdiv(GNN_E * (GNN_DHID / 4), TB), TB, 0, stream>>>(src, dst, norm, T, A, GNN_E, GNN_N);
    }
    // --- layer 2: T = relu(A) @ W2 ; A = agg(T) + b2 ---
    {
        int tiles = (GNN_N / 16) * (GNN_DHID / 16);
        gnn_gemm_wmma<GNN_DHID, GNN_DHID, true><<<cdiv(tiles, 8), 256, 0, stream>>>(A, W2, T, GNN_N);
        gnn_agg_init<GNN_DHID><<<cdiv(GNN_N * (GNN_DHID / 4), TB), TB, 0, stream>>>(A, b2, T, dinv, GNN_N);
        gnn_agg_edges<GNN_DHID><<<cdiv(GNN_E * (GNN_DHID / 4), TB), TB, 0, stream>>>(src, dst, norm, T, A, GNN_E, GNN_N);
    }
    // --- layer 3: T = relu(A) @ W3 ; out = agg(T) + b3 ---
    {
        int tiles = (GNN_N / 16) * (GNN_DOUT / 16);
        gnn_gemm_wmma<GNN_DHID, GNN_DOUT, true><<<cdiv(tiles, 8), 256, 0, stream>>>(A, W3, T, GNN_N);
        gnn_agg_init<GNN_DOUT><<<cdiv(GNN_N * (GNN_DOUT / 4), TB), TB, 0, stream>>>(out, b3, T, dinv, GNN_N);
        gnn_agg_edges<GNN_DOUT><<<cdiv(GNN_E * (GNN_DOUT / 4), TB), TB, 0, stream>>>(src, dst, norm, T, out, GNN_E, GNN_N);
    }
}